// ItemVectorTransform_67276367725290
// MI455X (gfx1250) — compile-verified
//
#include <hip/hip_runtime.h>

// MI455X / gfx1250, wave32. Fused kernel for:
//   r = exp(keys @ u); u_ = sum_i r_i * values_i; out = [u ; u_]
// HBM-bound (409.6 MB streamed, ~0.5 flop/byte): one pass over keys+values,
// fp32 WMMA (16x16x4) for the 16-row dot-product tiles, VALU for the
// weighted value accumulation, LDS + global f32 atomics for the reduction.

typedef __attribute__((ext_vector_type(2))) float v2f;
typedef __attribute__((ext_vector_type(4))) float v4f;
typedef __attribute__((ext_vector_type(8))) float v8f;

#define D_DIM 256
#define TILE_M 16
#define BLOCK_THREADS 256
#define WAVES_PER_BLOCK (BLOCK_THREADS / 32)
#define GRID_BLOCKS 256

__global__ void __launch_bounds__(BLOCK_THREADS)
iv_init_kernel(const float* __restrict__ u, float* __restrict__ out) {
    const int tid = threadIdx.x;        // 256 threads, one block
    out[tid] = u[tid];                  // out[0:256]   = u
    out[D_DIM + tid] = 0.0f;            // out[256:512] = 0 (accumulated below)
}

__global__ void __launch_bounds__(BLOCK_THREADS)
iv_kv_exp_kernel(const float* __restrict__ u,
                 const float* __restrict__ keys,
                 const float* __restrict__ values,
                 float* __restrict__ out, int M) {
    __shared__ float s_u[D_DIM];
    __shared__ float s_acc[D_DIM];

    const int tid  = threadIdx.x;
    const int lane = tid & 31;
    const int wave = tid >> 5;

    s_u[tid]   = u[tid];
    s_acc[tid] = 0.0f;
    __syncthreads();

    // WMMA f32 16x16x4 operand mapping (ISA 7.12.2, 32-bit layouts):
    //   A 16x4: v0 = {K0 lanes0-15 | K2 lanes16-31}, v1 = {K1 | K3}
    //   B 4x16: v0 = {K0 | K2},                     v1 = {K1 | K3}
    // Lane L<16 handles key-row (L) columns (k, k+1); lane L>=16 handles
    // the same row's columns (k+2, k+3).
    const int koff  = (lane & 16) ? 2 : 0;
    const int rowIn = lane & 15;

    // B = u replicated across all 16 output columns; loop-invariant, so
    // hoist all 64 K-chunks (128 VGPRs) out of the tile loop.
    v2f bmat[D_DIM / 4];
#pragma unroll
    for (int kb = 0; kb < D_DIM / 4; ++kb) {
        bmat[kb].x = s_u[kb * 4 + koff];
        bmat[kb].y = s_u[kb * 4 + koff + 1];
    }

    // Per-lane accumulator for u_ columns [lane*8, lane*8+8)
    v4f acc0 = {0.f, 0.f, 0.f, 0.f};
    v4f acc1 = {0.f, 0.f, 0.f, 0.f};

    const int nTiles = M / TILE_M;                       // 12500
    const int nWaves = gridDim.x * WAVES_PER_BLOCK;
    const int gwave  = blockIdx.x * WAVES_PER_BLOCK + wave;

    for (int t = gwave; t < nTiles; t += nWaves) {
        const int row0 = t * TILE_M;
        const float* arow = keys + (size_t)(row0 + rowIn) * D_DIM + koff;

        // Prefetch the start of each of the 16 key rows of this wave's next tile.
        if (t + nWaves < nTiles) {
            __builtin_prefetch(
                keys + (size_t)(row0 + nWaves * TILE_M + rowIn) * D_DIM, 0, 0);
        }

        // dots[i] = keys[row0+i] . u  via 64 chained f32 WMMAs
        v8f c = {};
#pragma unroll
        for (int kb = 0; kb < D_DIM / 4; ++kb) {
            v2f a = *(const v2f*)(arow + kb * 4);
            c = __builtin_amdgcn_wmma_f32_16x16x4_f32(
                    /*neg_a=*/false, a, /*neg_b=*/false, bmat[kb],
                    /*c_mod=*/(short)0, c, /*reuse_a=*/false, /*reuse_b=*/false);
        }

        // C[i][j] is the row-i dot replicated over j:
        //   rows 0..7  -> c[i]   on lanes 0..15
        //   rows 8..15 -> c[i-8] on lanes 16..31
        // Broadcast via constant-lane shuffle, exp, then accumulate values.
        const float* vrow = values + (size_t)row0 * D_DIM + lane * 8;
#pragma unroll
        for (int i = 0; i < TILE_M; ++i) {
            float dv = __shfl(c[i & 7], (i & 8) ? 16 : 0, 32);
            float rv = __expf(dv);
            v4f va = __builtin_nontemporal_load(
                         (const v4f*)(vrow + (size_t)i * D_DIM));
            v4f vb = __builtin_nontemporal_load(
                         (const v4f*)(vrow + (size_t)i * D_DIM + 4));
            acc0 += va * rv;
            acc1 += vb * rv;
        }
    }

    // Block-level reduction: LDS f32 atomics (lanes hit distinct addresses
    // within a wave; waves contend benignly), then one global atomic per
    // column per block.
    const int colBase = lane * 8;
#pragma unroll
    for (int q = 0; q < 4; ++q) atomicAdd(&s_acc[colBase + q],     acc0[q]);
#pragma unroll
    for (int q = 0; q < 4; ++q) atomicAdd(&s_acc[colBase + 4 + q], acc1[q]);
    __syncthreads();

    atomicAdd(&out[D_DIM + tid], s_acc[tid]);
}

extern "C" void kernel_launch(void* const* d_in, const int* in_sizes, int n_in,
                              void* d_out, int out_size, void* d_ws, size_t ws_size,
                              hipStream_t stream) {
    (void)n_in; (void)out_size; (void)d_ws; (void)ws_size;
    const float* u      = (const float*)d_in[0];   // (256,)
    const float* keys   = (const float*)d_in[1];   // (M, 256)
    const float* values = (const float*)d_in[2];   // (M, 256)
    float* out = (float*)d_out;                    // (2, 256, 1) flat = 512 floats

    const int M = in_sizes[1] / D_DIM;             // 200000

    iv_init_kernel<<<1, BLOCK_THREADS, 0, stream>>>(u, out);
    iv_kv_exp_kernel<<<GRID_BLOCKS, BLOCK_THREADS, 0, stream>>>(u, keys, values,
                                                                out, M);
}